// GraphVToS_6451040878946
// MI455X (gfx1250) — compile-verified
//
#include <hip/hip_runtime.h>

// ---------------------------------------------------------------------------
// GraphVToS on MI455X (gfx1250, wave32).
//
// Phase 1 (wmma_gemm_kernel): fp32 WMMA GEMM
//   a_buf[row, k] = sum_f vf[row, f] * w_vs[f,      k]
//   c_buf[row, k] = sum_f vf[row, f] * w_vs[128+f,  k]
//   where row = (b*N + i)*COOR + c, treated as one GEMM with 256 output cols.
//   Uses V_WMMA_F32_16X16X4_F32, one 16x16 tile per wave, K accumulated 4/step.
//
// Phase 2 (pair_kernel): bandwidth-bound pairwise combine, coalesced stores:
//   out[b,i,j,k] = sum_c d[b,i,j,c] * (a[b,i,c,k] + c[b,j,c,k] + bias[k])
// ---------------------------------------------------------------------------

typedef float v2f __attribute__((ext_vector_type(2)));
typedef float v8f __attribute__((ext_vector_type(8)));

#define B_    8
#define N_    128
#define COOR_ 3
#define F_    128
#define FILT_ 128
#define MROWS (B_ * N_ * COOR_)   // 3072 rows

// ---- Phase 1: fused dual GEMM via f32 WMMA --------------------------------
// Tiles: 192 M-tiles x 16 N-tiles (256 combined columns) = 3072 tiles,
// one wave each. blockDim=256 (8 waves), gridDim=384.
__global__ __launch_bounds__(256) void wmma_gemm_kernel(
    const float* __restrict__ vf,   // [3072, 128]
    const float* __restrict__ w,    // [256, 128]  (w_vs)
    float* __restrict__ a_buf,      // [3072, 128]
    float* __restrict__ c_buf) {    // [3072, 128]
  const int wave = (blockIdx.x * blockDim.x + threadIdx.x) >> 5;
  const int lane = threadIdx.x & 31;

  const int mt = wave >> 4;          // 0..191
  const int nt = wave & 15;          // 0..15
  const int m0 = mt * 16;
  const int n0 = nt * 16;            // column in combined 256-wide output

  const int lmod  = lane & 15;
  const int hi    = lane >> 4;       // 0 for lanes 0-15, 1 for 16-31
  const int khalf = hi << 1;         // K sub-offset: 0 or 2

  const int arow    = m0 + lmod;           // A-matrix row for this lane
  const int ncol    = n0 + lmod;           // B/D column (0..255 combined)
  const int wcol    = ncol & 127;          // column within w_vs [*,128]
  const int wrowoff = (ncol >> 7) << 7;    // +128 rows for the j-part weights

  v8f acc = {};
  #pragma unroll 4
  for (int k0 = 0; k0 < F_; k0 += 4) {
    const int ka = k0 + khalf;
    v2f a, b;
    // A 16x4 fp32 fragment: lanes 0-15 hold K={k0,k0+1}, lanes 16-31 K={k0+2,k0+3}
    a.x = vf[arow * F_ + ka];
    a.y = vf[arow * F_ + ka + 1];
    // B 4x16 fp32 fragment: same K split, column = lane%16
    b.x = w[(wrowoff + ka)     * FILT_ + wcol];
    b.y = w[(wrowoff + ka + 1) * FILT_ + wcol];
    // (neg_a, A, neg_b, B, c_mod, C, reuse_a, reuse_b)
    acc = __builtin_amdgcn_wmma_f32_16x16x4_f32(
        false, a, false, b, (short)0, acc, false, false);
  }

  // D 16x16 fp32: VGPR v holds row m0 + v + 8*hi, column = lane%16.
  float* __restrict__ dst = (ncol < FILT_) ? a_buf : c_buf;
  #pragma unroll
  for (int v = 0; v < 8; ++v) {
    const int m = m0 + v + (hi << 3);
    dst[m * FILT_ + wcol] = acc[v];
  }
}

// ---- Phase 2: pairwise combine, store-bandwidth bound ---------------------
// One block per (b,i): 128 threads (thread = k), loop over j.
// Stores are contiguous 512B per j -> saturates HBM write path.
__global__ __launch_bounds__(128) void pair_kernel(
    const float* __restrict__ a_buf,   // [B*N, 3, 128]
    const float* __restrict__ c_buf,   // [B*N, 3, 128]
    const float* __restrict__ dist,    // [B, N, N, 3]
    const float* __restrict__ bias,    // [128]
    float* __restrict__ out) {         // [B, N, N, 128]
  const int bi = blockIdx.x;           // (b*N + i)
  const int b  = bi >> 7;
  const int k  = threadIdx.x;

  // i-side projections: reused across all 128 j's -> keep in registers.
  const float a0 = a_buf[(bi * 3 + 0) * FILT_ + k];
  const float a1 = a_buf[(bi * 3 + 1) * FILT_ + k];
  const float a2 = a_buf[(bi * 3 + 2) * FILT_ + k];
  const float bv = bias[k];

  const float* __restrict__ drow = dist  + (size_t)bi * N_ * COOR_;
  const float* __restrict__ crow = c_buf + (size_t)b  * N_ * COOR_ * FILT_;
  float*       __restrict__ orow = out   + (size_t)bi * N_ * FILT_;

  for (int j = 0; j < N_; ++j) {
    // Uniform across the block: one cacheline, broadcast by HW.
    const float d0 = drow[j * 3 + 0];
    const float d1 = drow[j * 3 + 1];
    const float d2 = drow[j * 3 + 2];
    // j-side projections: coalesced 512B loads, L2-resident (3 MB total).
    const float c0 = crow[(j * 3 + 0) * FILT_ + k];
    const float c1 = crow[(j * 3 + 1) * FILT_ + k];
    const float c2 = crow[(j * 3 + 2) * FILT_ + k];
    const float r = fmaf(d0, a0 + c0 + bv,
                    fmaf(d1, a1 + c1 + bv,
                         d2 * (a2 + c2 + bv)));
    orow[j * FILT_ + k] = r;
  }
}

extern "C" void kernel_launch(void* const* d_in, const int* in_sizes, int n_in,
                              void* d_out, int out_size, void* d_ws, size_t ws_size,
                              hipStream_t stream) {
  const float* vf   = (const float*)d_in[0];  // [B,N,COOR,F]
  const float* dist = (const float*)d_in[1];  // [B,N,N,COOR]
  const float* w    = (const float*)d_in[2];  // [2F, FILT]
  const float* bvs  = (const float*)d_in[3];  // [FILT]
  float* outp  = (float*)d_out;               // [B,N,N,FILT]

  float* a_buf = (float*)d_ws;                         // 3072*128 floats
  float* c_buf = a_buf + (size_t)MROWS * FILT_;        // 3072*128 floats
  (void)in_sizes; (void)n_in; (void)out_size; (void)ws_size;

  // Phase 1: 3072 tiles / 8 waves per block = 384 blocks.
  wmma_gemm_kernel<<<384, 256, 0, stream>>>(vf, w, a_buf, c_buf);
  // Phase 2: one block per (b,i).
  pair_kernel<<<B_ * N_, 128, 0, stream>>>(a_buf, c_buf, dist, bvs, outp);
}